// AdaptiveDepth2normal_56684978373070
// MI455X (gfx1250) — compile-verified
//
#include <hip/hip_runtime.h>
#include <math.h>

#define BATCH   2
#define HH      384
#define WW      512
#define HWSZ    (HH*WW)
#define KS      5
#define K2      (KS*KS)               // 25
#define TILE_X  32
#define TILE_Y  8
#define HALO_X  (TILE_X + 4)          // 36
#define HALO_Y  (TILE_Y + 4)          // 12
#define HALO_N  (HALO_X * HALO_Y)     // 432
#define GW_TILE (TILE_X * TILE_Y * K2)      // 6400 floats
#define GW_ROW  (TILE_X * K2)               // 800 floats per image row chunk
#define GW_ROW4 (GW_ROW / 4)                // 200 float4 per chunk
#define GW_N4   (GW_TILE / 4)               // 1600 float4 per tile
#define DEPTH_MAX_F 10.0f

// ---------------------------------------------------------------------------
// CDNA5 async global -> LDS copies (ASYNCcnt-tracked).
// ---------------------------------------------------------------------------
__device__ __forceinline__ void async_load_b32_to_lds(const float* gptr,
                                                      unsigned lds_byte_off) {
  asm volatile("global_load_async_to_lds_b32 %0, %1, off"
               :: "v"(lds_byte_off), "v"(gptr)
               : "memory");
}

__device__ __forceinline__ void async_load_b128_to_lds(const float4* gptr,
                                                       unsigned lds_byte_off) {
  asm volatile("global_load_async_to_lds_b128 %0, %1, off"
               :: "v"(lds_byte_off), "v"(gptr)
               : "memory");
}

__device__ __forceinline__ void wait_async_zero() {
#if __has_builtin(__builtin_amdgcn_s_wait_asynccnt)
  __builtin_amdgcn_s_wait_asynccnt(0);
#else
  asm volatile("s_wait_asynccnt 0" ::: "memory");
#endif
}

__device__ __forceinline__ float valid_of(float d) {
  return (d > 0.0f && d < DEPTH_MAX_F) ? 1.0f : 0.0f;
}

// ---------------------------------------------------------------------------
// Kernel 1: points = K^-1 @ [x, y, 1] * depth  -> written to d_out 2nd half.
// b is uniform per block so intrinsic loads / inverse stay on the SALU path.
// ---------------------------------------------------------------------------
__global__ __launch_bounds__(256) void points_kernel(
    const float* __restrict__ depth, const float* __restrict__ intr,
    float* __restrict__ pts)
{
  const int b = blockIdx.y;
  const int idx = blockIdx.x * 256 + threadIdx.x;
  if (idx >= HWSZ) return;
  const int y = idx / WW;
  const int x = idx - y * WW;

  const float* Kb = intr + b * 9;
  const float a00 = Kb[0], a01 = Kb[1], a02 = Kb[2];
  const float a10 = Kb[3], a11 = Kb[4], a12 = Kb[5];
  const float a20 = Kb[6], a21 = Kb[7], a22 = Kb[8];

  const float c00 =  (a11*a22 - a12*a21);
  const float c01 = -(a10*a22 - a12*a20);
  const float c02 =  (a10*a21 - a11*a20);
  const float invdet = 1.0f / (a00*c00 + a01*c01 + a02*c02);

  const float i00 =  (a11*a22 - a12*a21) * invdet;
  const float i01 = -(a01*a22 - a02*a21) * invdet;
  const float i02 =  (a01*a12 - a02*a11) * invdet;
  const float i10 = -(a10*a22 - a12*a20) * invdet;
  const float i11 =  (a00*a22 - a02*a20) * invdet;
  const float i12 = -(a00*a12 - a02*a10) * invdet;
  const float i20 =  (a10*a21 - a11*a20) * invdet;
  const float i21 = -(a00*a21 - a01*a20) * invdet;
  const float i22 =  (a00*a11 - a01*a10) * invdet;

  const float d  = depth[(size_t)b * HWSZ + idx];
  const float fx = (float)x, fy = (float)y;

  const size_t base = (size_t)b * 3 * HWSZ + (size_t)y * WW + x;
  pts[base]            = (i00*fx + i01*fy + i02) * d;
  pts[base +   HWSZ]   = (i10*fx + i11*fy + i12) * d;
  pts[base + 2*HWSZ]   = (i20*fx + i21*fy + i22) * d;
}

// ---------------------------------------------------------------------------
// Kernel 2: adaptive normals.
//  * 32x8 pixel tile + 2-texel halo of (px,py,pz,depth) float4 in LDS,
//    filled with async b32 copies from the planar points / depth buffers.
//  * guide weights for the tile staged as [pixel][25] in LDS via fully
//    coalesced async b128 copies (8 contiguous 800-float chunks).
//  * inner loop: 3x ds_load_b128 point gathers + 3x conflict-free
//    ds_load_b32 guide reads per triangle; single-pass softmax.
// ---------------------------------------------------------------------------
__global__ __launch_bounds__(256) void normals_kernel(
    const float* __restrict__ depth, const float* __restrict__ guide,
    const int*   __restrict__ tri,   const float* __restrict__ areaw,
    const float* __restrict__ pts,   float* __restrict__ out, int T)
{
  __shared__ float4 sP[HALO_N];     // 6912 B
  __shared__ float  sGw[GW_TILE];   // 25600 B

  const int b  = blockIdx.z;
  const int x0 = blockIdx.x * TILE_X;
  const int y0 = blockIdx.y * TILE_Y;
  const int lx = threadIdx.x, ly = threadIdx.y;
  const int tid = ly * TILE_X + lx;

  // Stage guide weights: row r of the tile is one contiguous 800-float chunk
  // both in global memory and in the [pixel][25] LDS layout.
  for (int e4 = tid; e4 < GW_N4; e4 += 256) {
    const int r  = e4 / GW_ROW4;
    const int j4 = e4 - r * GW_ROW4;
    const float* gsrc =
        guide + (((size_t)b * HH + (y0 + r)) * WW + x0) * K2;
    async_load_b128_to_lds((const float4*)gsrc + j4,
                           (unsigned)(size_t)(&sGw[r * GW_ROW + j4 * 4]));
  }

  // Zero-fill halo: reference zero-pads points and validity outside image.
  for (int tt = tid; tt < HALO_N; tt += 256)
    sP[tt] = make_float4(0.0f, 0.0f, 0.0f, 0.0f);
  __syncthreads();   // zero-fill visible before async halo writes

  const float* pb = pts   + (size_t)b * 3 * HWSZ;
  const float* db = depth + (size_t)b * HWSZ;
  for (int tt = tid; tt < HALO_N; tt += 256) {
    const int hy = tt / HALO_X;
    const int hx = tt - hy * HALO_X;
    const int gx = x0 - 2 + hx;
    const int gy = y0 - 2 + hy;
    if (gx >= 0 && gx < WW && gy >= 0 && gy < HH) {
      const unsigned loff = (unsigned)(size_t)(&sP[tt]);
      const size_t gidx = (size_t)gy * WW + gx;
      async_load_b32_to_lds(pb + gidx,            loff + 0u);   // px
      async_load_b32_to_lds(pb + HWSZ + gidx,     loff + 4u);   // py
      async_load_b32_to_lds(pb + 2*HWSZ + gidx,   loff + 8u);   // pz
      async_load_b32_to_lds(db + gidx,            loff + 12u);  // depth
    }
  }
  wait_async_zero();   // my wave's async LDS writes landed
  __syncthreads();     // everyone's landed

  const float4 Pc = sP[(ly + 2) * HALO_X + (lx + 2)];
  const float centerValid = valid_of(Pc.w);
  const int gwbase = tid * K2;

  float accx = 0.0f, accy = 0.0f, accz = 0.0f, accS = 0.0f;
  for (int t = 0; t < T; ++t) {
    // Triangle metadata is wave-uniform -> scalar loads.
    const int ia = tri[3*t + 0];
    const int ib = tri[3*t + 1];
    const int ic = tri[3*t + 2];
    const float aw = areaw[t];

    const int o1 = (ly + ia / KS) * HALO_X + (lx + ia % KS);
    const int o2 = (ly + ib / KS) * HALO_X + (lx + ib % KS);
    const int o3 = (ly + ic / KS) * HALO_X + (lx + ic % KS);

    const float4 P1 = sP[o1];   // ds_load_b128
    const float4 P2 = sP[o2];
    const float4 P3 = sP[o3];

    const float e1x = P2.x - P1.x, e1y = P2.y - P1.y, e1z = P2.z - P1.z;
    const float e2x = P3.x - P1.x, e2y = P3.y - P1.y, e2z = P3.z - P1.z;

    float nx = e1y * e2z - e1z * e2y;
    float ny = e1z * e2x - e1x * e2z;
    float nz = e1x * e2y - e1y * e2x;
    const float inl = 1.0f / (sqrtf(nx*nx + ny*ny + nz*nz) + 1e-5f);
    nx *= inl; ny *= inl; nz *= inl;

    const float v = valid_of(P1.w) * valid_of(P2.w) * valid_of(P3.w);
    // stride 25 dwords across lanes: 25 coprime with 64 banks -> no conflict
    const float g = sGw[gwbase + ia] * sGw[gwbase + ib] * sGw[gwbase + ic];
    // fw in [0,1] -> softmax is safe without max-subtraction.
    const float w = __expf(v * aw * g);

    accS += w;
    accx += w * nx; accy += w * ny; accz += w * nz;
  }

  const float is = 1.0f / accS;          // accS >= T >= 1
  accx *= is; accy *= is; accz *= is;
  const float sc = centerValid /
      (sqrtf(accx*accx + accy*accy + accz*accz) + 1e-5f);

  const int x = x0 + lx, y = y0 + ly;
  const size_t obase = ((size_t)b * 3 * HH + y) * WW + x;
  out[obase]            = accx * sc;
  out[obase +   HWSZ]   = accy * sc;
  out[obase + 2*HWSZ]   = accz * sc;
}

// ---------------------------------------------------------------------------
extern "C" void kernel_launch(void* const* d_in, const int* in_sizes, int n_in,
                              void* d_out, int out_size, void* d_ws, size_t ws_size,
                              hipStream_t stream) {
  const float* depth = (const float*)d_in[0];
  const float* intr  = (const float*)d_in[1];
  const float* guide = (const float*)d_in[2];
  const int*   tri   = (const int*)  d_in[3];
  const float* aw    = (const float*)d_in[4];
  const int T = in_sizes[3] / 3;

  float* out = (float*)d_out;
  float* pts = out + (size_t)BATCH * 3 * HWSZ;   // second output = staging buf

  points_kernel<<<dim3((HWSZ + 255) / 256, BATCH), 256, 0, stream>>>(
      depth, intr, pts);

  normals_kernel<<<dim3(WW / TILE_X, HH / TILE_Y, BATCH), dim3(TILE_X, TILE_Y),
                   0, stream>>>(depth, guide, tri, aw, pts, out, T);
}